// DifferentiableDLT_72971494359097
// MI455X (gfx1250) — compile-verified
//
#include <hip/hip_runtime.h>
#include <math.h>

typedef float v2f __attribute__((ext_vector_type(2)));
typedef float v8f __attribute__((ext_vector_type(8)));

#define NPTS   484      // int(sqrt(512))^2 = 22*22
#define NGRID  22
#define HF     315
#define WF     560
#define NCHUNK 242      // 2*NPTS rows / 4 (K=4 per WMMA)
#define EPS_DLT 1e-6f

__device__ __forceinline__ float block_reduce_sum(float v, float* red, int tid) {
    red[tid] = v; __syncthreads();
    #pragma unroll
    for (int s = 128; s > 0; s >>= 1) {
        if (tid < s) red[tid] += red[tid + s];
        __syncthreads();
    }
    float r = red[0]; __syncthreads();
    return r;
}

__device__ __forceinline__ float block_reduce_max(float v, float* red, int tid) {
    red[tid] = v; __syncthreads();
    #pragma unroll
    for (int s = 128; s > 0; s >>= 1) {
        if (tid < s) red[tid] = fmaxf(red[tid], red[tid + s]);
        __syncthreads();
    }
    float r = red[0]; __syncthreads();
    return r;
}

__global__ __launch_bounds__(256)
void dlt_homography_kernel(const float* __restrict__ flow,
                           const float* __restrict__ mask,
                           const int*   __restrict__ img_h_p,
                           const int*   __restrict__ img_w_p,
                           float*       __restrict__ out) {
    const int b   = blockIdx.x;
    const int tid = threadIdx.x;

    // point data: (sx, sy, dx, dy) packed for ds_load_b128, weight separate
    __shared__ float4 ptv[NPTS];
    __shared__ float  wv[NPTS];
    __shared__ float  red[256];
    __shared__ float  Cpart[8 * 256];   // per-wave 16x16 Gram partials
    __shared__ float  Msh[256];         // reduced 16x16 (valid 9x9)
    __shared__ float  Mj[81], Vj[81], hv[9];

    const float scale_x = (float)img_w_p[0] / (float)WF;   // 1120/560 = 2
    const float scale_y = (float)img_h_p[0] / (float)HF;   // 630/315  = 2

    // ---------------- Phase 1: bilinear gather at 22x22 grid ----------------
    const size_t fbase = (size_t)b * 2 * HF * WF;
    const size_t mbase = (size_t)b * HF * WF;
    for (int p = tid; p < NPTS; p += 256) {
        const int gi = p / NGRID, gj = p - gi * NGRID;
        const float x = (float)(28.0 + (double)gj * (503.0 / 21.0)); // linspace(28,531,22)
        const float y = (float)(15.0 + (double)gi * (284.0 / 21.0)); // linspace(15,299,22)
        const float x0 = floorf(x), y0 = floorf(y);
        const float wx = x - x0,   wy = y - y0;
        int ix0 = (int)x0, iy0 = (int)y0;
        int ix1 = min(ix0 + 1, WF - 1), iy1 = min(iy0 + 1, HF - 1);
        ix0 = max(ix0, 0); iy0 = max(iy0, 0);
        const float w00 = (1.f-wx)*(1.f-wy), w10 = wx*(1.f-wy);
        const float w01 = (1.f-wx)*wy,       w11 = wx*wy;
        const size_t r0 = (size_t)iy0 * WF, r1 = (size_t)iy1 * WF;
        const float* f0 = flow + fbase;                       // flow channel 0 (x)
        const float* f1 = flow + fbase + (size_t)HF * WF;     // flow channel 1 (y)
        const float* mm = mask + mbase;
        const float fx = f0[r0+ix0]*w00 + f0[r0+ix1]*w10 + f0[r1+ix0]*w01 + f0[r1+ix1]*w11;
        const float fy = f1[r0+ix0]*w00 + f1[r0+ix1]*w10 + f1[r1+ix0]*w01 + f1[r1+ix1]*w11;
        const float wm = mm[r0+ix0]*w00 + mm[r0+ix1]*w10 + mm[r1+ix0]*w01 + mm[r1+ix1]*w11;
        ptv[p] = make_float4(x * scale_x, y * scale_y,
                             (x + fx) * scale_x, (y + fy) * scale_y);
        wv[p]  = wm * 10.0f;
    }
    __syncthreads();

    // ---------------- Phase 2: softmax weights + Hartley normalization -------
    const int p0 = tid, p1 = tid + 256;
    const bool h1 = (p1 < NPTS);

    float l0 = wv[p0];
    float l1 = h1 ? wv[p1] : -1e30f;
    const float wmax = block_reduce_max(fmaxf(l0, l1), red, tid);
    const float e0 = expf(l0 - wmax);
    const float e1 = h1 ? expf(l1 - wmax) : 0.f;
    const float esum = block_reduce_sum(e0 + e1, red, tid);
    const float inv_esum = 1.0f / esum;
    wv[p0] = e0 * inv_esum;
    if (h1) wv[p1] = e1 * inv_esum;

    float4 P0 = ptv[p0];
    float4 P1 = h1 ? ptv[p1] : make_float4(0.f, 0.f, 0.f, 0.f);

    // means (unweighted, as in reference)
    const float invN = 1.0f / (float)NPTS;
    const float mxs = block_reduce_sum(P0.x + P1.x, red, tid) * invN;
    const float mys = block_reduce_sum(P0.y + P1.y, red, tid) * invN;
    const float mxd = block_reduce_sum(P0.z + P1.z, red, tid) * invN;
    const float myd = block_reduce_sum(P0.w + P1.w, red, tid) * invN;

    // mean centered norms -> scale
    float ns = 0.f, nd = 0.f;
    {
        float cx = P0.x-mxs, cy = P0.y-mys; ns += sqrtf(cx*cx+cy*cy);
        cx = P0.z-mxd; cy = P0.w-myd;       nd += sqrtf(cx*cx+cy*cy);
        if (h1) {
            cx = P1.x-mxs; cy = P1.y-mys; ns += sqrtf(cx*cx+cy*cy);
            cx = P1.z-mxd; cy = P1.w-myd; nd += sqrtf(cx*cx+cy*cy);
        }
    }
    const float scale_s = fmaxf(block_reduce_sum(ns, red, tid) * invN * 0.70710678118654752f, 1e-8f);
    const float scale_d = fmaxf(block_reduce_sum(nd, red, tid) * invN * 0.70710678118654752f, 1e-8f);
    const float invs = 1.0f / scale_s;
    const float invd = 1.0f / scale_d;

    // normalize points in place
    ptv[p0] = make_float4((P0.x-mxs)*invs, (P0.y-mys)*invs,
                          (P0.z-mxd)*invd, (P0.w-myd)*invd);
    if (h1)
        ptv[p1] = make_float4((P1.x-mxs)*invs, (P1.y-mys)*invs,
                              (P1.z-mxd)*invd, (P1.w-myd)*invd);
    __syncthreads();

    // ---------------- Phase 3: M = A^T A via V_WMMA_F32_16X16X4_F32 ----------
    // Chunk t covers 4 rows of A: point 2t (rows r=0,1) and point 2t+1 (rows r=0,1).
    // A-operand 16x4: lane m=lane&15 holds K-pair 2*(lane>>4) in VGPR0/1.
    // B-operand 4x16: lane n=lane&15 holds the same K-pair striping.
    // -> symmetric Gram: both operands identical per lane.
    //
    // Row values for column jc are a fixed linear form in (x, y, 1, xp|yp*{x,y,1}):
    //   r==0: [-x,-y,-1, 0,0,0, xp*x, xp*y, xp]
    //   r==1: [ 0, 0, 0,-x,-y,-1, yp*x, yp*y, yp]
    // Hoist the column selection into per-lane 0/±1 coefficients (branchless loop).
    const int wid  = tid >> 5;
    const int lane = tid & 31;
    const int hlf  = lane >> 4;
    const int jc   = lane & 15;

    const float cAx0 = (jc == 0) ? -1.f : 0.f;
    const float cAy0 = (jc == 1) ? -1.f : 0.f;
    const float cAc0 = (jc == 2) ? -1.f : 0.f;
    const float cAx1 = (jc == 3) ? -1.f : 0.f;
    const float cAy1 = (jc == 4) ? -1.f : 0.f;
    const float cAc1 = (jc == 5) ? -1.f : 0.f;
    const float cPx  = (jc == 6) ?  1.f : 0.f;
    const float cPy  = (jc == 7) ?  1.f : 0.f;
    const float cPc  = (jc == 8) ?  1.f : 0.f;

    v8f c = {0.f, 0.f, 0.f, 0.f, 0.f, 0.f, 0.f, 0.f};
    #pragma unroll 2
    for (int t = wid; t < NCHUNK; t += 8) {
        const int pt = 2 * t + hlf;
        const float4 P = ptv[pt];           // ds_load_b128 (broadcast per half-wave)
        const float  w = wv[pt];            // ds_load_b32
        const float  q = cPx * P.x + cPy * P.y + cPc;   // x/y/1 selector for cols 6..8
        v2f a;
        a.x = w * (cAx0 * P.x + cAy0 * P.y + cAc0 + P.z * q);   // row r=0 (uses xp)
        a.y = w * (cAx1 * P.x + cAy1 * P.y + cAc1 + P.w * q);   // row r=1 (uses yp)
        c = __builtin_amdgcn_wmma_f32_16x16x4_f32(false, a, false, a,
                                                  (short)0, c, false, false);
    }
    #pragma unroll
    for (int r = 0; r < 8; ++r)
        Cpart[wid * 256 + (r + hlf * 8) * 16 + jc] = c[r];
    __syncthreads();

    {   // reduce 8 wave partials; add eps*I on the 9x9 block
        float s = 0.f;
        #pragma unroll
        for (int w8 = 0; w8 < 8; ++w8) s += Cpart[w8 * 256 + tid];
        const int row = tid >> 4, col = tid & 15;
        if (row == col && row < 9) s += EPS_DLT;
        Msh[tid] = s;
    }
    __syncthreads();

    // ---------------- Phase 4+5: Jacobi eigensolve + denormalize (thread 0) --
    if (tid == 0) {
        for (int i = 0; i < 9; ++i)
            for (int k = 0; k < 9; ++k) {
                Mj[i*9+k] = Msh[i*16+k];
                Vj[i*9+k] = (i == k) ? 1.f : 0.f;
            }
        for (int sweep = 0; sweep < 12; ++sweep) {
            for (int p = 0; p < 8; ++p)
                for (int q = p + 1; q < 9; ++q) {
                    const float apq = Mj[p*9+q];
                    if (fabsf(apq) < 1e-24f) continue;
                    const float theta = (Mj[q*9+q] - Mj[p*9+p]) / (2.f * apq);
                    float tt = 1.f / (fabsf(theta) + sqrtf(theta*theta + 1.f));
                    if (theta < 0.f) tt = -tt;
                    const float cc = 1.f / sqrtf(tt*tt + 1.f);
                    const float ss = tt * cc;
                    for (int i = 0; i < 9; ++i) {          // A <- A*J
                        const float aip = Mj[i*9+p], aiq = Mj[i*9+q];
                        Mj[i*9+p] = cc*aip - ss*aiq;
                        Mj[i*9+q] = ss*aip + cc*aiq;
                    }
                    for (int i = 0; i < 9; ++i) {          // A <- J^T*A
                        const float api = Mj[p*9+i], aqi = Mj[q*9+i];
                        Mj[p*9+i] = cc*api - ss*aqi;
                        Mj[q*9+i] = ss*api + cc*aqi;
                    }
                    for (int i = 0; i < 9; ++i) {          // V <- V*J
                        const float vip = Vj[i*9+p], viq = Vj[i*9+q];
                        Vj[i*9+p] = cc*vip - ss*viq;
                        Vj[i*9+q] = ss*vip + cc*viq;
                    }
                }
        }
        int   qm   = 0;
        float dmin = Mj[0];
        for (int k = 1; k < 9; ++k) {
            const float d = Mj[k*9+k];
            if (d < dmin) { dmin = d; qm = k; }
        }
        const float sgn = (Vj[8*9+qm] < 0.f) ? -1.f : 1.f;
        #pragma unroll
        for (int k = 0; k < 9; ++k) hv[k] = sgn * Vj[k*9+qm];

        // H = T_dst^-1 * Hn * T_src ; T_src = [[is,0,-mxs*is],[0,is,-mys*is],[0,0,1]]
        // T_dst^-1 = [[scale_d,0,mxd],[0,scale_d,myd],[0,0,1]]
        const float is = invs, txs = -mxs * invs, tys = -mys * invs;
        float G[9], Hm[9];
        #pragma unroll
        for (int i = 0; i < 3; ++i) {
            G[i*3+0] = hv[i*3+0] * is;
            G[i*3+1] = hv[i*3+1] * is;
            G[i*3+2] = hv[i*3+0]*txs + hv[i*3+1]*tys + hv[i*3+2];
        }
        #pragma unroll
        for (int jj = 0; jj < 3; ++jj) {
            Hm[0*3+jj] = scale_d * G[0*3+jj] + mxd * G[2*3+jj];
            Hm[1*3+jj] = scale_d * G[1*3+jj] + myd * G[2*3+jj];
            Hm[2*3+jj] = G[2*3+jj];
        }
        const float isf = 1.0f / fmaxf(Hm[8], 1e-8f);
        #pragma unroll
        for (int k = 0; k < 9; ++k) out[(size_t)b * 9 + k] = Hm[k] * isf;
    }
}

extern "C" void kernel_launch(void* const* d_in, const int* in_sizes, int n_in,
                              void* d_out, int out_size, void* d_ws, size_t ws_size,
                              hipStream_t stream) {
    const float* flow  = (const float*)d_in[0];
    const float* mask  = (const float*)d_in[1];
    const int*   img_h = (const int*)d_in[2];
    const int*   img_w = (const int*)d_in[3];
    float* out = (float*)d_out;
    const int B = in_sizes[0] / (2 * HF * WF);
    dlt_homography_kernel<<<dim3(B), dim3(256), 0, stream>>>(flow, mask, img_h, img_w, out);
}